// HyperbolicGraphConvolution_17317308137938
// MI455X (gfx1250) — compile-verified
//
#include <hip/hip_runtime.h>
#include <cstdint>
#include <cstddef>

#define DFEAT 65
#define PADF  66   // padded row stride for internal buffers: 66*4 = 264 B, 8B-aligned
#define TB    256

#if defined(__AMDGCN__) && __has_builtin(__builtin_amdgcn_global_load_async_to_lds_b32) && __has_builtin(__builtin_amdgcn_s_wait_asynccnt)
#define HAVE_ASYNC_LDS 1
typedef __attribute__((address_space(1))) int g_int;  // global AS pointer-to-int
typedef __attribute__((address_space(3))) int l_int;  // LDS AS pointer-to-int
#else
#define HAVE_ASYNC_LDS 0
#endif

// ---------- helpers ----------
__device__ __forceinline__ float wave_reduce_sum(float v) {
  // wave32 butterfly; result uniform across all 32 lanes
  for (int m = 16; m >= 1; m >>= 1) v += __shfl_xor(v, m, 32);
  return v;
}

// ---------- CSR build ----------
__global__ void zero_ints_k(int* __restrict__ p, int n) {
  int i = blockIdx.x * blockDim.x + threadIdx.x;
  if (i < n) p[i] = 0;
}

__global__ void count_k(const int* __restrict__ rows, int* __restrict__ cnt, int e) {
  int i = blockIdx.x * blockDim.x + threadIdx.x;
  if (i < e) atomicAdd(&cnt[rows[i]], 1);
}

// single-block exclusive scan over n counts -> offs[0..n]
__global__ void scan_k(const int* __restrict__ cnt, int* __restrict__ offs, int n) {
  __shared__ int sh[1024];
  int t = threadIdx.x;
  int chunk = (n + 1023) >> 10;
  int b = t * chunk;
  int e = min(b + chunk, n);
  int s = 0;
  for (int i = b; i < e; ++i) s += cnt[i];
  sh[t] = s;
  __syncthreads();
  for (int off = 1; off < 1024; off <<= 1) {
    int v = (t >= off) ? sh[t - off] : 0;
    __syncthreads();
    sh[t] += v;
    __syncthreads();
  }
  int run = (t == 0) ? 0 : sh[t - 1];
  for (int i = b; i < e; ++i) { offs[i] = run; run += cnt[i]; }
  if (t == 1023) offs[n] = sh[1023];
}

__global__ void fill_k(const int* __restrict__ rows, const int* __restrict__ cols,
                       const float* __restrict__ vals, const int* __restrict__ offs,
                       int* __restrict__ cursor, int* __restrict__ colp,
                       float* __restrict__ valp, int e) {
  int i = blockIdx.x * blockDim.x + threadIdx.x;
  if (i < e) {
    int r = rows[i];
    int p = offs[r] + atomicAdd(&cursor[r], 1);
    colp[p] = cols[i];
    valp[p] = vals[i];
  }
}

// ---------- logmap0 (write padded h) + zero the accumulator (d_out, stride 65) ----------
__global__ __launch_bounds__(TB) void logmap_k(const float* __restrict__ x,
                                               float* __restrict__ h,
                                               float* __restrict__ acc, int n) {
  int gtid = blockIdx.x * blockDim.x + threadIdx.x;
  int wave = gtid >> 5;
  int lane = gtid & 31;
  if (wave >= n) return;
  size_t bx = (size_t)wave * DFEAT;   // input / acc stride 65
  size_t bh = (size_t)wave * PADF;    // padded stride 66
  float y0 = x[bx + 1 + lane];        // features 1..32
  float y1 = x[bx + 33 + lane];       // features 33..64
  float sq = wave_reduce_sum(y0 * y0 + y1 * y1);
  float ynorm = fmaxf(sqrtf(sq), 1e-15f);
  float x0 = x[bx];                   // broadcast load
  float theta = fmaxf(x0, 1.0f + 1e-7f);
  float ach = logf(theta + sqrtf(theta * theta - 1.0f));  // arccosh
  float fac = ach / ynorm;
  h[bh + 1 + lane]  = fac * y0;
  h[bh + 33 + lane] = fac * y1;
  if (lane == 0) h[bh] = 0.0f;        // tangent first component = 0 (invariant under SpMM)
  acc[bx + lane]      = 0.0f;
  acc[bx + 32 + lane] = 0.0f;
  if (lane == 0) acc[bx + 64] = 0.0f;
}

// ---------- pull-mode SpMM (padded layout): one wave per row, fused acc += ----------
// Per edge: 1x global_load_b64 feature-pair gather + 1 broadcast b32 + 1 ds_load_b64.
__global__ __launch_bounds__(TB) void spmm_pull_k(const int* __restrict__ offs,
                                                  const int* __restrict__ colp,
                                                  const float* __restrict__ valp,
                                                  const float* __restrict__ hin,
                                                  float* __restrict__ hout,
                                                  float* __restrict__ acc, int n) {
  int gtid = blockIdx.x * blockDim.x + threadIdx.x;
  int wave = gtid >> 5;
  int lane = gtid & 31;
  if (wave >= n) return;
  int row = __builtin_amdgcn_readfirstlane(wave);  // SGPR row -> scalar offs loads
  int s = offs[row];
  int e = offs[row + 1];
  __builtin_prefetch(colp + s, 0, 3);   // global_prefetch_b8 on gfx1250
  __builtin_prefetch(valp + s, 0, 3);
  float a0 = 0.0f, a1 = 0.0f, a2 = 0.0f;  // features 2*lane, 2*lane+1, 64
#if HAVE_ASYNC_LDS
  __shared__ int2 se[TB];               // interleaved {col, val-bits}, 32 edges/wave
  int wslot = (threadIdx.x >> 5) << 5;
  for (int base = s; base < e; base += 32) {
    int p = base + lane;
    if (p < e) {
      __builtin_amdgcn_global_load_async_to_lds_b32(
          (g_int*)(colp + p), (l_int*)&se[wslot + lane].x, 0, 0);
      __builtin_amdgcn_global_load_async_to_lds_b32(
          (g_int*)(valp + p), (l_int*)&se[wslot + lane].y, 0, 0);
    }
    __builtin_amdgcn_s_wait_asynccnt(0);
    int m = min(32, e - base);
    for (int j = 0; j < m; ++j) {
      int2  cv = se[wslot + j];         // single ds_load_b64 broadcast
      int   c  = cv.x;
      float v  = __int_as_float(cv.y);
      const float* src = hin + (size_t)c * PADF;
      float2 hv = ((const float2*)src)[lane];   // aligned b64 gather, features 2l,2l+1
      a0 = fmaf(v, hv.x, a0);
      a1 = fmaf(v, hv.y, a1);
      a2 = fmaf(v, src[64], a2);                // broadcast feature 64
    }
  }
#else
  for (int base = s; base < e; base += 32) {
    int p = base + lane;
    int c = 0; float v = 0.0f;
    if (p < e) { c = colp[p]; v = valp[p]; }    // coalesced tile stage in registers
    int m = min(32, e - base);
    for (int j = 0; j < m; ++j) {
      int   cc = __shfl(c, j, 32);
      float vv = __shfl(v, j, 32);
      const float* src = hin + (size_t)cc * PADF;
      float2 hv = ((const float2*)src)[lane];
      a0 = fmaf(vv, hv.x, a0);
      a1 = fmaf(vv, hv.y, a1);
      a2 = fmaf(vv, src[64], a2);
    }
  }
#endif
  float* dst = hout + (size_t)row * PADF;
  float2 o; o.x = a0; o.y = a1;
  ((float2*)dst)[lane] = o;             // aligned b64 store, features 2l,2l+1
  if (lane == 0) dst[64] = a2;
  float* ad = acc + (size_t)row * DFEAT;  // this wave owns the row: plain RMW (per-row, cheap)
  ad[2 * lane]     += a0;
  ad[2 * lane + 1] += a1;
  if (lane == 0) ad[64] += a2;
}

// ---------- expmap0 + proj (in-place on acc == d_out, stride 65) ----------
__global__ __launch_bounds__(TB) void expproj_k(float* __restrict__ io, int n) {
  int gtid = blockIdx.x * blockDim.x + threadIdx.x;
  int wave = gtid >> 5;
  int lane = gtid & 31;
  if (wave >= n) return;
  size_t b = (size_t)wave * DFEAT;
  float u0 = io[b + 2 * lane];        // feature 0 (lane 0) is identically 0
  float u1 = io[b + 2 * lane + 1];
  float u64 = io[b + 64];             // broadcast
  float sq = wave_reduce_sum(u0 * u0 + u1 * u1) + u64 * u64;
  float xn = fmaxf(sqrtf(sq), 1e-15f);
  float sh = sinhf(xn);
  float fac = sh / xn;
  float t0 = fac * u0;
  float t1 = fac * u1;
  float t64 = fac * u64;
  float tsq = wave_reduce_sum(t0 * t0 + t1 * t1) + t64 * t64;
  float first = sqrtf(fmaxf(1.0f + tsq, 1e-7f));
  if (lane == 0) {
    io[b]     = first;                // overwrite feature 0 with proj first component
    io[b + 1] = t1;
    io[b + 64] = t64;
  } else {
    io[b + 2 * lane]     = t0;
    io[b + 2 * lane + 1] = t1;
  }
}

extern "C" void kernel_launch(void* const* d_in, const int* in_sizes, int n_in,
                              void* d_out, int out_size, void* d_ws, size_t ws_size,
                              hipStream_t stream) {
  const float* x    = (const float*)d_in[0];
  const int*   rows = (const int*)d_in[1];
  const int*   cols = (const int*)d_in[2];
  const float* vals = (const float*)d_in[3];
  int n = in_sizes[0] / DFEAT;
  int e = in_sizes[1];
  float* out = (float*)d_out;
  if (n <= 0 || e <= 0) return;

  // workspace layout (padded feature buffers)
  float* ping   = (float*)d_ws;                       // n*66
  float* pong   = ping + (size_t)n * PADF;            // n*66
  int*   offs   = (int*)(pong + (size_t)n * PADF);    // n+1
  int*   cursor = offs + (n + 1);                     // n
  int*   colp   = cursor + n;                         // e
  float* valp   = (float*)(colp + e);                 // e
  size_t need = (size_t)((char*)(valp + e) - (char*)d_ws);
  if (ws_size < need) return;

  int eb = (e + TB - 1) / TB;
  int nb = (n + TB - 1) / TB;
  int rowblocks = (int)(((size_t)n * 32 + TB - 1) / TB);  // one wave32 per row

  // CSR build (reused by all 3 layers)
  zero_ints_k<<<nb, TB, 0, stream>>>(cursor, n);
  count_k<<<eb, TB, 0, stream>>>(rows, cursor, e);
  scan_k<<<1, 1024, 0, stream>>>(cursor, offs, n);
  zero_ints_k<<<nb, TB, 0, stream>>>(cursor, n);
  fill_k<<<eb, TB, 0, stream>>>(rows, cols, vals, offs, cursor, colp, valp, e);

  // tangent map + zero accumulator (d_out)
  logmap_k<<<rowblocks, TB, 0, stream>>>(x, ping, out, n);

  // three GCN layers, acc fused into pull kernel
  spmm_pull_k<<<rowblocks, TB, 0, stream>>>(offs, colp, valp, ping, pong, out, n);
  spmm_pull_k<<<rowblocks, TB, 0, stream>>>(offs, colp, valp, pong, ping, out, n);
  spmm_pull_k<<<rowblocks, TB, 0, stream>>>(offs, colp, valp, ping, pong, out, n);

  // expmap0 + proj, in place on d_out
  expproj_k<<<rowblocks, TB, 0, stream>>>(out, n);
}